// GCNConv_85693187490237
// MI455X (gfx1250) — compile-verified
//
#include <hip/hip_runtime.h>
#include <hip/hip_bf16.h>

typedef __attribute__((ext_vector_type(2))) float v2f;
typedef __attribute__((ext_vector_type(8))) float v8f;

#define F_IN  128
#define F_OUT 256

// ---------------------------------------------------------------------------
// Zero a dword range of workspace (bitmap + partial-sum vector), grid-stride.
// ---------------------------------------------------------------------------
__global__ void k_zero_u32(unsigned int* __restrict__ p, long long n) {
  long long i = (long long)blockIdx.x * blockDim.x + threadIdx.x;
  long long stride = (long long)gridDim.x * blockDim.x;
  for (; i < n; i += stride) p[i] = 0u;
}

// ---------------------------------------------------------------------------
// Set-semantics adjacency: bit (u,v) and (v,u) in an N x ceil(N/32) bitmap.
// Duplicate edges collapse exactly like adj.at[...].set(1.0) in the reference.
// ---------------------------------------------------------------------------
__global__ void k_scatter_bits(const int* __restrict__ ei,
                               unsigned int* __restrict__ bm,
                               int E, int W32) {
  int t = blockIdx.x * blockDim.x + threadIdx.x;
  if (t >= E) return;
  int u = ei[t];
  int v = ei[E + t];
  atomicOr(&bm[(size_t)u * W32 + (v >> 5)], 1u << (v & 31));
  atomicOr(&bm[(size_t)v * W32 + (u >> 5)], 1u << (u & 31));
}

// ---------------------------------------------------------------------------
// deg[i] = popcount(row i of bitmap); dinv[i] = 1/sqrt(deg). One wave32/node.
// ---------------------------------------------------------------------------
__global__ void k_degree_bits(const unsigned int* __restrict__ bm,
                              float* __restrict__ dinv, int N, int W32) {
  int node = blockIdx.x * (blockDim.x >> 5) + (threadIdx.x >> 5);
  int lane = threadIdx.x & 31;
  if (node >= N) return;  // whole-wave uniform exit
  const unsigned int* row = bm + (size_t)node * W32;
  int cnt = 0;
  for (int c = lane; c < W32; c += 32) cnt += __popc(row[c]);
#pragma unroll
  for (int off = 16; off > 0; off >>= 1) cnt += __shfl_xor(cnt, off, 32);
  if (lane == 0) dinv[node] = 1.0f / sqrtf((float)cnt);
}

// ---------- fallback path (ws too small for bitmap): additive degree --------
__global__ void k_scatter_deg(const int* __restrict__ ei, int* __restrict__ deg,
                              int E) {
  int t = blockIdx.x * blockDim.x + threadIdx.x;
  if (t >= E) return;
  int u = ei[t];
  int v = ei[E + t];
  if (u == v) {
    atomicAdd(&deg[u], 1);
  } else {
    atomicAdd(&deg[u], 1);
    atomicAdd(&deg[v], 1);
  }
}
__global__ void k_dinv_from_deg(const int* __restrict__ deg,
                                float* __restrict__ dinv, int N) {
  int i = blockIdx.x * blockDim.x + threadIdx.x;
  if (i < N) dinv[i] = 1.0f / sqrtf((float)deg[i]);
}

// ---------------------------------------------------------------------------
// s[k] = sum_i dinv[i] * x[i,k]   (k = 0..127). blockDim.x == 128.
// Rank-1 identity: dinv @ (x @ W) == (dinv @ x) @ W, so we reduce x first
// (6.1 MB read) instead of materializing h = x@W (12.3 MB write + read).
// ---------------------------------------------------------------------------
__global__ void k_reduce_s(const float* __restrict__ x,
                           const float* __restrict__ dinv,
                           float* __restrict__ s, int N) {
  int f = threadIdx.x;  // feature 0..127
  float acc = 0.0f;
  for (int i = blockIdx.x; i < N; i += gridDim.x)
    acc += dinv[i] * x[(size_t)i * F_IN + f];
  atomicAdd(&s[f], acc);
}

// ---------------------------------------------------------------------------
// agg[f] = sum_k s[k] * W[k,f]. Single block of 256 threads; W is 128 KB.
// ---------------------------------------------------------------------------
__global__ void k_agg(const float* __restrict__ s, const float* __restrict__ w,
                      float* __restrict__ agg) {
  int f = threadIdx.x;  // 0..255
  float acc = 0.0f;
#pragma unroll 8
  for (int k = 0; k < F_IN; ++k) acc += s[k] * w[k * F_OUT + f];
  agg[f] = acc;
}

// ---------------------------------------------------------------------------
// out[i,f] = dinv[i]*agg[f] + bias[f] via V_WMMA_F32_16X16X4_F32.
// One wave per 16x16 output tile. A: 16x4 with only K=0 column = dinv tile
// (doc layout: lanes 0-15 hold M=0..15, VGPR0 = K=0). B: 4x16 with only K=0
// row = agg tile (lanes 0-15 = N=0..15 in VGPR0). All other K slots zero, so
// D = outer(dinv_tile, agg_tile) + C exactly, with C = bias broadcast along M.
//
// Loads are unconditional (clamped index) + cndmask-select, so no EXEC
// branches before the WMMA. Full tiles (the only kind when N%16==0) use a
// single base pointer with constant immediate store offsets -> 8 back-to-back
// global_store_b32, no per-store guards.
// ---------------------------------------------------------------------------
__global__ void k_outer_wmma(const float* __restrict__ dinv,
                             const float* __restrict__ agg,
                             const float* __restrict__ bias,
                             float* __restrict__ out, int N, int nTiles) {
  const int wave = (int)((blockIdx.x * blockDim.x + threadIdx.x) >> 5);
  if (wave >= nTiles) return;            // uniform per-wave exit
  const int lane    = threadIdx.x & 31;
  const int rowTile = wave >> 4;         // 16 column tiles per row strip
  const int colTile = wave & 15;
  const int row0    = rowTile * 16;
  const int col0    = colTile * 16;
  const int l15     = lane & 15;
  const bool lo     = lane < 16;

  int rIdx = row0 + l15;
  if (rIdx >= N) rIdx = N - 1;           // clamp so every lane loads validly

  // Unconditional loads for all 32 lanes, then select -> no EXEC branching.
  const float dv = dinv[rIdx];
  const float av = agg[col0 + l15];

  v2f a;
  a.x = lo ? dv : 0.0f;                  // A[M=l15, K=0]
  a.y = 0.0f;                            // K=1 (and lanes>=16: K=2,3) are zero
  v2f b;
  b.x = lo ? av : 0.0f;                  // B[K=0, N=l15]
  b.y = 0.0f;

  const float bv = bias[col0 + l15];     // C[m, n=l15] = bias[n] for all m
  v8f c;
#pragma unroll
  for (int r = 0; r < 8; ++r) c[r] = bv;

  v8f d = __builtin_amdgcn_wmma_f32_16x16x4_f32(
      /*neg_a=*/false, a, /*neg_b=*/false, b,
      /*c_mod=*/(short)0, c, /*reuse_a=*/false, /*reuse_b=*/false);

  // D layout: VGPR r -> M = r + (lane>=16 ? 8 : 0), N = lane&15.
  const int rbase = row0 + (lo ? 0 : 8);
  float* p = out + (size_t)rbase * F_OUT + (col0 + l15);

  if (row0 + 16 <= N) {
    // Full tile: one address, 8 stores with compile-time byte offsets.
#pragma unroll
    for (int r = 0; r < 8; ++r) p[(size_t)r * F_OUT] = d[r];
  } else {
    // Ragged last tile (unused when N%16==0, kept for generality).
#pragma unroll
    for (int r = 0; r < 8; ++r)
      if (rbase + r < N) p[(size_t)r * F_OUT] = d[r];
  }
}

// ---------------------------------------------------------------------------
extern "C" void kernel_launch(void* const* d_in, const int* in_sizes, int n_in,
                              void* d_out, int out_size, void* d_ws,
                              size_t ws_size, hipStream_t stream) {
  const float* x    = (const float*)d_in[0];
  const int*   ei   = (const int*)d_in[1];
  const float* w    = (const float*)d_in[2];
  const float* bias = (const float*)d_in[3];
  float*       out  = (float*)d_out;

  const int N   = in_sizes[0] / F_IN;   // 12000
  const int E   = in_sizes[1] / 2;      // 384000
  const int W32 = (N + 31) / 32;        // 375 dwords per bitmap row

  const size_t bmDw     = (size_t)N * (size_t)W32;          // bitmap dwords
  const size_t needFull = (bmDw + 128 + F_OUT + N) * 4;     // bm + s + agg + dinv

  char* ws = (char*)d_ws;
  const bool fullPath = (ws_size >= needFull);

  float* dinv;
  float* s;
  float* aggv;

  if (fullPath) {
    unsigned int* bm = (unsigned int*)ws;
    s    = (float*)(ws + bmDw * 4);                  // contiguous with bitmap
    aggv = (float*)(ws + (bmDw + 128) * 4);
    dinv = (float*)(ws + (bmDw + 128 + F_OUT) * 4);

    // 1. zero bitmap + s in one pass
    k_zero_u32<<<2048, 256, 0, stream>>>(bm, (long long)(bmDw + 128));
    // 2. scatter adjacency bits (set semantics => duplicates collapse)
    k_scatter_bits<<<(E + 255) / 256, 256, 0, stream>>>(ei, bm, E, W32);
    // 3. dinv = 1/sqrt(popcount(row))
    k_degree_bits<<<(N + 7) / 8, 256, 0, stream>>>(bm, dinv, N, W32);
  } else {
    int* deg = (int*)ws;
    s    = (float*)(ws + (size_t)N * 4);
    aggv = (float*)(ws + ((size_t)N + 128) * 4);
    dinv = (float*)(ws + ((size_t)N + 128 + F_OUT) * 4);

    k_zero_u32<<<256, 256, 0, stream>>>((unsigned int*)deg, (long long)(N + 128));
    k_scatter_deg<<<(E + 255) / 256, 256, 0, stream>>>(ei, deg, E);
    k_dinv_from_deg<<<(N + 255) / 256, 256, 0, stream>>>(deg, dinv, N);
  }

  // 4. s = dinv @ x   (length-128)
  k_reduce_s<<<240, 128, 0, stream>>>(x, dinv, s, N);
  // 5. agg = s @ W    (length-256)
  k_agg<<<1, 256, 0, stream>>>(s, w, aggv);
  // 6. out = outer(dinv, agg) + bias  -- one WMMA per 16x16 tile
  const int rowTiles = (N + 15) / 16;
  const int nTiles   = rowTiles * (F_OUT / 16);      // waves needed
  const int blocks   = (nTiles + 7) / 8;             // 8 waves per 256-thr block
  k_outer_wmma<<<blocks, 256, 0, stream>>>(dinv, aggv, bias, out, N, nTiles);
}